// GraphConvLayer_70506183131139
// MI455X (gfx1250) — compile-verified
//
#include <hip/hip_runtime.h>
#include <hip/hip_bf16.h>

// Problem constants (from reference)
#define BB   16
#define NN   2048
#define EE   65536
#define DIN  256
#define DOUT 256
#define KTOT 512   // 2*DIN

typedef float v2f __attribute__((ext_vector_type(2)));
typedef float v8f __attribute__((ext_vector_type(8)));

// Hardware FP32 atomic add (non-returning global_atomic_add_f32).
// Workspace is coarse-grained device memory, so the "unsafe" HW atomic is valid.
__device__ __forceinline__ void hw_atomic_add(float* addr, float v) {
    unsafeAtomicAdd(addr, v);
}

// ---------------------------------------------------------------------------
// Kernel 1: zero the workspace (sums + counts). Must run every launch because
// scatter accumulates and the harness replays the graph without re-poisoning.
// ---------------------------------------------------------------------------
__global__ void gc_zero_ws(float4* __restrict__ ws4, int n4) {
    int i = blockIdx.x * blockDim.x + threadIdx.x;
    if (i < n4) {
        ws4[i] = make_float4(0.f, 0.f, 0.f, 0.f);
    }
}

// ---------------------------------------------------------------------------
// Kernel 2: edge scatter. One wave32 per edge; lane l handles columns
// l, l+32, ..., l+224 so every atomic instruction is a coalesced 32-lane
// burst over consecutive floats. node_features fits in L2 (33.5MB << 192MB),
// so gathers are L2 hits and the f32 atomics resolve in L2.
// ---------------------------------------------------------------------------
__global__ void gc_scatter(const float* __restrict__ feats,
                           const long long* __restrict__ edges,
                           float* __restrict__ sums,
                           float* __restrict__ counts) {
    const int lane = threadIdx.x & 31;
    const int widx = threadIdx.x >> 5;           // 8 waves per 256-thread block
    const int gid  = blockIdx.x * 8 + widx;      // global edge slot: 0 .. B*E-1
    const int b    = gid >> 16;                  // E = 65536
    const int e    = gid & (EE - 1);

    const long long src = edges[(size_t)b * 2 * EE + e];
    const long long dst = edges[(size_t)b * 2 * EE + EE + e];

    const float* __restrict__ frow = feats + ((size_t)b * NN + (size_t)src) * DIN;
    float*       __restrict__ srow = sums  + ((size_t)b * NN + (size_t)dst) * DIN;

#pragma unroll
    for (int i = 0; i < 8; ++i) {
        const int c = lane + 32 * i;
        hw_atomic_add(&srow[c], frow[c]);
    }
    if (lane == 0) {
        hw_atomic_add(&counts[b * NN + (int)dst], 1.0f);
    }
}

// ---------------------------------------------------------------------------
// Kernel 3: normalize sums -> mean (divide by max(count,1)), in place.
// ---------------------------------------------------------------------------
__global__ void gc_normalize(float4* __restrict__ sums4,
                             const float* __restrict__ counts, int n4) {
    int i = blockIdx.x * blockDim.x + threadIdx.x;
    if (i >= n4) return;
    const int bn = i >> 6;                       // 64 float4 per 256-wide row
    const float c = counts[bn];
    const float inv = 1.0f / fmaxf(c, 1.0f);
    float4 v = sums4[i];
    v.x *= inv; v.y *= inv; v.z *= inv; v.w *= inv;
    sums4[i] = v;
}

// ---------------------------------------------------------------------------
// Kernel 4: WMMA f32 GEMM.
//   out[r, o] = sum_{k<256} feats[r,k]*W[o,k] + sum_{k<256} agg[r,k]*W[o,256+k] + bias[o]
// Block: 256 threads = 8 waves arranged 4 (M) x 2 (N).
// Wave tile: 32 rows x 128 cols = 2x8 tiles of 16x16, acc in 128 VGPRs.
// K loop in steps of 4 using V_WMMA_F32_16X16X4_F32 (full f32 precision,
// matching the f32 reference; bf16 A/B would inject ~0.4% error at K=512).
//
// ISA fragment layouts (32-bit):
//   A 16x4 : lane<16 -> {K=k0,k0+1} of row M=lane ; lane>=16 -> {K=k0+2,k0+3}
//   B 4x16 : lane<16 -> {K=k0,k0+1} of col N=lane ; lane>=16 -> {K=k0+2,k0+3}
//   C/D    : VGPR r, lane -> M = r + 8*(lane>>4), N = lane&15
// ---------------------------------------------------------------------------
__global__ void __launch_bounds__(256)
gc_gemm_wmma(const float* __restrict__ feats,   // (B*N, 256) first half of K
             const float* __restrict__ agg,     // (B*N, 256) second half of K
             const float* __restrict__ W,       // (256, 512)
             const float* __restrict__ bias,    // (256,)
             float* __restrict__ out) {         // (B*N, 256)
    const int lane = threadIdx.x & 31;
    const int wave = threadIdx.x >> 5;
    const int wm   = wave >> 1;                  // 0..3
    const int wn   = wave & 1;                   // 0..1
    const int half = lane >> 4;                  // 0 or 1 -> K sub-pair
    const int l15  = lane & 15;

    const int rowBase = blockIdx.x * 128 + wm * 32;  // covers 2 M-tiles
    const int colBase = wn * 128;                    // covers 8 N-tiles

    v8f acc[2][8];
#pragma unroll
    for (int mt = 0; mt < 2; ++mt)
#pragma unroll
        for (int nt = 0; nt < 8; ++nt)
            acc[mt][nt] = (v8f)(0.0f);

    // per-lane base offsets
    const size_t arow0 = (size_t)(rowBase + l15) * DIN;        // M tile 0 row
    const size_t arow1 = (size_t)(rowBase + 16 + l15) * DIN;   // M tile 1 row

    for (int k0 = 0; k0 < KTOT; k0 += 4) {
        const int kk = k0 + half * 2;            // this lane's K pair start
        const int khalf = kk >> 8;               // 0: feats, 1: agg
        const int koff = kk & (DIN - 1);
        const float* __restrict__ Asrc = khalf ? agg : feats;

        const v2f a0 = *reinterpret_cast<const v2f*>(Asrc + arow0 + koff);
        const v2f a1 = *reinterpret_cast<const v2f*>(Asrc + arow1 + koff);

        v2f bfrag[8];
#pragma unroll
        for (int nt = 0; nt < 8; ++nt) {
            const int col = colBase + nt * 16 + l15;
            bfrag[nt] = *reinterpret_cast<const v2f*>(W + (size_t)col * KTOT + kk);
        }

#pragma unroll
        for (int nt = 0; nt < 8; ++nt) {
            acc[0][nt] = __builtin_amdgcn_wmma_f32_16x16x4_f32(
                false, a0, false, bfrag[nt], (short)0, acc[0][nt], false, false);
            acc[1][nt] = __builtin_amdgcn_wmma_f32_16x16x4_f32(
                false, a1, false, bfrag[nt], (short)0, acc[1][nt], false, false);
        }
    }

    // Epilogue: bias add + store per documented C/D layout.
#pragma unroll
    for (int mt = 0; mt < 2; ++mt) {
        const int mtile = rowBase + mt * 16 + half * 8;
#pragma unroll
        for (int nt = 0; nt < 8; ++nt) {
            const int col = colBase + nt * 16 + l15;
            const float bv = bias[col];
#pragma unroll
            for (int r = 0; r < 8; ++r) {
                out[(size_t)(mtile + r) * DOUT + col] = acc[mt][nt][r] + bv;
            }
        }
    }
}

// ---------------------------------------------------------------------------
// Host-side launcher
// ---------------------------------------------------------------------------
extern "C" void kernel_launch(void* const* d_in, const int* in_sizes, int n_in,
                              void* d_out, int out_size, void* d_ws, size_t ws_size,
                              hipStream_t stream) {
    const float*     feats = (const float*)d_in[0];      // (16, 2048, 256)
    const long long* edges = (const long long*)d_in[1];  // (16, 2, 65536) int64
    const float*     W     = (const float*)d_in[2];      // (256, 512)
    const float*     bias  = (const float*)d_in[3];      // (256,)
    float*           out   = (float*)d_out;              // (16, 2048, 256)

    float* sums   = (float*)d_ws;                        // B*N*DIN floats
    float* counts = sums + (size_t)BB * NN * DIN;        // B*N floats

    // 1) zero sums + counts
    {
        const int nfloats = BB * NN * DIN + BB * NN;     // 8,421,376 (div by 4)
        const int n4 = nfloats / 4;
        gc_zero_ws<<<(n4 + 255) / 256, 256, 0, stream>>>((float4*)d_ws, n4);
    }

    // 2) scatter-add per edge (wave per edge)
    {
        const int nblocks = (BB * EE) / 8;               // 131072
        gc_scatter<<<nblocks, 256, 0, stream>>>(feats, edges, sums, counts);
    }

    // 3) normalize to mean
    {
        const int n4 = (BB * NN * DIN) / 4;              // 2,097,152
        gc_normalize<<<(n4 + 255) / 256, 256, 0, stream>>>((float4*)sums, counts, n4);
    }

    // 4) WMMA GEMM + bias
    {
        const int nblocks = (BB * NN) / 128;             // 256
        gc_gemm_wmma<<<nblocks, 256, 0, stream>>>(feats, sums, W, bias, out);
    }
}